// Aggregator_41686952575440
// MI455X (gfx1250) — compile-verified
//
#include <hip/hip_runtime.h>
#include <hip/hip_bf16.h>

// MI455X / gfx1250, wave32. WMMA fp32 16x16x4 GEMM fused with channel reduction.

typedef float v2f __attribute__((ext_vector_type(2)));
typedef float v4f __attribute__((ext_vector_type(4)));
typedef float v8f __attribute__((ext_vector_type(8)));

#define NUM_VIEWS 24
#define P_DIM 196          // 14*14
#define OUT_DIM 1024
#define W_COLS 392         // 2*P
#define SLOPE 0.2f
#define MBLOCK 32          // batch rows per block
#define BATCH 16384

// ---------------------------------------------------------------------------
// Setup kernel: cnt[c], ii_sum[p], const[o] = Wi[o,:]·ii_sum + 24*b_aggre[o].
// ws layout (floats): [0..1023] const vector, [1024..1047] cnt.
// grid = 4 blocks x 256 threads.
// ---------------------------------------------------------------------------
__global__ __launch_bounds__(256) void aggre_setup(
    const int* __restrict__ idx, const float* __restrict__ W_idx,
    const float* __restrict__ b_idx, const float* __restrict__ Wagg,
    const float* __restrict__ b_aggre, float* __restrict__ ws) {
  __shared__ float ii[P_DIM];
  const int t = threadIdx.x;

  if (blockIdx.x == 0 && t < NUM_VIEWS) {
    int c = 0;
#pragma unroll
    for (int k = 0; k < NUM_VIEWS; ++k) c += (idx[k] == t) ? 1 : 0;
    ws[OUT_DIM + t] = (float)c;
  }

  if (t < P_DIM) {
    const float bi = b_idx[t];
    float sum = 0.f;
#pragma unroll
    for (int k = 0; k < NUM_VIEWS; ++k) {
      float v = W_idx[t * NUM_VIEWS + idx[k]] + bi;
      sum += (v > 0.f) ? v : SLOPE * v;
    }
    ii[t] = sum;
  }
  __syncthreads();

  const int o = blockIdx.x * 256 + t;
  float acc = (float)NUM_VIEWS * b_aggre[o];
  const float* wrow = Wagg + (long long)o * W_COLS + P_DIM;  // Wi row
#pragma unroll 4
  for (int p = 0; p < P_DIM; ++p) acc += wrow[p] * ii[p];
  ws[o] = acc;
}

// ---------------------------------------------------------------------------
// Main kernel: per block (256 thr = 8 waves), 32 batch rows.
// Phase 1: s[r,p] = sum_c cnt[c]*x[b0+r,p,c]  -> LDS (stride 196 floats).
// Phase 2: out[b, :] = s @ Wx^T + const  via v_wmma_f32_16x16x4_f32.
//   wave w: M-tile = (w&1)*16 rows, N-strip = (w>>1)*256 .. +256.
// grid = BATCH/32 = 512 blocks.
// ---------------------------------------------------------------------------
__global__ __launch_bounds__(256) void aggre_main(
    const float* __restrict__ x, const float* __restrict__ Wagg,
    const float* __restrict__ ws, float* __restrict__ out) {
  __shared__ float s_lds[MBLOCK * P_DIM];

  const int t = threadIdx.x;
  const int b0 = blockIdx.x * MBLOCK;

  // channel weights (broadcast load, hot in cache)
  float cnt[NUM_VIEWS];
#pragma unroll
  for (int c = 0; c < NUM_VIEWS; ++c) cnt[c] = ws[OUT_DIM + c];

  // ---- Phase 1: streaming channel reduction (x is HBM-streamed once; NT) ----
  for (int i = t; i < MBLOCK * P_DIM; i += 256) {
    const int r = i / P_DIM;
    const int p = i - r * P_DIM;
    const v4f* xv =
        (const v4f*)(x + ((long long)(b0 + r) * P_DIM + p) * NUM_VIEWS);
    __builtin_prefetch(xv + 6 * 256, 0, 0);  // next grid-stride iteration
    float sum = 0.f;
#pragma unroll
    for (int q = 0; q < 6; ++q) {            // 24 floats = 6x float4
      v4f v = __builtin_nontemporal_load(&xv[q]);
      sum += cnt[4 * q + 0] * v.x + cnt[4 * q + 1] * v.y +
             cnt[4 * q + 2] * v.z + cnt[4 * q + 3] * v.w;
    }
    s_lds[i] = sum;
  }
  __syncthreads();

  // ---- Phase 2: WMMA GEMM (uniform control flow; EXEC all ones) ----
  const int wave  = t >> 5;       // wave32
  const int lane  = t & 31;
  const int lhalf = lane & 15;
  const int khalf = lane >> 4;    // 0: K={0,1} / 1: K={2,3} (A); B mirrored
  const int mrow  = (wave & 1) * 16;
  const int nbase = (wave >> 1) * 256;

  // A fragment source: s_lds[(mrow+lhalf)*196 + khalf*2 + p0]  (ds_load_b64,
  // bank-conflict-free: row*196 % 64 steps by 4 over 16 rows)
  const int a_off = (mrow + lhalf) * P_DIM + khalf * 2;

  for (int g = 0; g < 4; ++g) {
    const int n0 = nbase + g * 64;
    v8f acc[4] = {};
    const float* bp[4];
#pragma unroll
    for (int tt = 0; tt < 4; ++tt)
      bp[tt] = Wagg + (long long)(n0 + tt * 16 + lhalf) * W_COLS + khalf * 2;

    for (int p0 = 0; p0 < P_DIM; p0 += 4) {
      v2f a = *(const v2f*)&s_lds[a_off + p0];
#pragma unroll
      for (int tt = 0; tt < 4; ++tt) {
        v2f b = *(const v2f*)(bp[tt] + p0);
        acc[tt] = __builtin_amdgcn_wmma_f32_16x16x4_f32(
            false, a, false, b, (short)0, acc[tt], false, false);
      }
    }

    // epilogue: + const[N], streamed NT store. lane -> N = n0+tt*16+lhalf,
    // vgpr r -> M = r + 8*khalf.
#pragma unroll
    for (int tt = 0; tt < 4; ++tt) {
      const int n = n0 + tt * 16 + lhalf;
      const float cv = ws[n];
      float* orow = out + (long long)(b0 + mrow + khalf * 8) * OUT_DIM + n;
#pragma unroll
      for (int r = 0; r < 8; ++r)
        __builtin_nontemporal_store(acc[tt][r] + cv, orow + (long long)r * OUT_DIM);
    }
  }
}

// ---------------------------------------------------------------------------
extern "C" void kernel_launch(void* const* d_in, const int* in_sizes, int n_in,
                              void* d_out, int out_size, void* d_ws, size_t ws_size,
                              hipStream_t stream) {
  const float* x       = (const float*)d_in[0];
  const int*   indices = (const int*)d_in[1];
  const float* W_idx   = (const float*)d_in[2];
  const float* b_idx   = (const float*)d_in[3];
  const float* W_aggre = (const float*)d_in[4];
  const float* b_aggre = (const float*)d_in[5];
  float* out = (float*)d_out;
  float* ws  = (float*)d_ws;   // needs (1024+24)*4 bytes

  aggre_setup<<<4, 256, 0, stream>>>(indices, W_idx, b_idx, W_aggre, b_aggre, ws);
  aggre_main<<<BATCH / MBLOCK, 256, 0, stream>>>(x, W_aggre, ws, out);
}